// SAGPools_75273596829856
// MI455X (gfx1250) — compile-verified
//
#include <hip/hip_runtime.h>
#include <hip/hip_bf16.h>

typedef __attribute__((ext_vector_type(2))) float v2f;
typedef __attribute__((ext_vector_type(4))) float v4f;
typedef __attribute__((ext_vector_type(8))) float v8f;

// ---------------------------------------------------------------------------
// Pass 1 & 3: GEMV  out[r] = sum_j M[r][j] * v[j]   (ncol fixed = 8192)
// One block = 256 threads = 8 waves, computes 16 output rows.
// Waves split the K dimension; tiles staged through padded LDS; the MAC is
// V_WMMA_F32_16X16X4_F32 with the vector in column 0 of B.
// ---------------------------------------------------------------------------
__global__ __launch_bounds__(256) void gemv16_wmma(const float* __restrict__ M,
                                                   const float* __restrict__ v,
                                                   float* __restrict__ out,
                                                   int nrows) {
    constexpr int NCOL = 8192;
    constexpr int TS   = 512;        // columns staged per round
    constexpr int LDT  = TS + 4;     // pad 4 floats -> conflict-free ds_read_b64
    __shared__ __align__(16) float tile[16 * LDT];   // ~33 KB
    __shared__ __align__(16) float vbuf[TS];         // 2 KB
    __shared__ float red[8][16];

    const int tid   = threadIdx.x;
    const int wave  = tid >> 5;
    const int lane  = tid & 31;
    const int row0  = blockIdx.x * 16;
    const int arow  = lane & 15;
    const int ahalf = lane >> 4;                     // 0: K=0..1, 1: K=2..3
    const bool col0 = (arow == 0);                   // lanes 0 and 16 feed B col 0

    v8f c = {};                                      // f32 16x16 accumulator

    for (int s = 0; s < NCOL / TS; ++s) {
        const int colbase = s * TS;
        __syncthreads();
        // ---- stage 16 x 512 f32 tile, fully coalesced b128 loads ----
        #pragma unroll
        for (int i = 0; i < (16 * TS / 4) / 256; ++i) {      // 8 float4 / thread
            int f  = tid + i * 256;
            int r  = f >> 7;                                  // / (TS/4)
            int c4 = f & 127;
            v4f g = *reinterpret_cast<const v4f*>(
                &M[(size_t)(row0 + r) * NCOL + colbase + c4 * 4]);
            *reinterpret_cast<v4f*>(&tile[r * LDT + c4 * 4]) = g;
        }
        vbuf[tid]       = v[colbase + tid];
        vbuf[tid + 256] = v[colbase + 256 + tid];
        __syncthreads();

        // ---- each wave: 64 columns of this stage, 16 WMMAs ----
        const int cw = wave * 64;
        #pragma unroll
        for (int k = 0; k < 64; k += 4) {
            const int kk = cw + k;
            // A fragment (16x4 f32): lane<16 -> K=0..1, lane>=16 -> K=2..3
            v2f A = *reinterpret_cast<const v2f*>(
                &tile[arow * LDT + kk + 2 * ahalf]);
            // B fragment (4x16 f32): only column N=0 nonzero = vector chunk
            v2f b2 = *reinterpret_cast<const v2f*>(&vbuf[kk + 2 * ahalf]);
            v2f B; B.x = col0 ? b2.x : 0.0f; B.y = col0 ? b2.y : 0.0f;
            c = __builtin_amdgcn_wmma_f32_16x16x4_f32(
                    false, A, false, B, (short)0, c, false, false);
        }
    }

    // D[m][0]: lane 0 holds rows 0..7 in c[0..7], lane 16 holds rows 8..15
    if (lane == 0) {
        #pragma unroll
        for (int i = 0; i < 8; ++i) red[wave][i] = c[i];
    }
    if (lane == 16) {
        #pragma unroll
        for (int i = 0; i < 8; ++i) red[wave][8 + i] = c[i];
    }
    __syncthreads();
    if (tid < 16) {
        float sum = 0.0f;
        #pragma unroll
        for (int w = 0; w < 8; ++w) sum += red[w][tid];
        if (row0 + tid < nrows) out[row0 + tid] = sum;
    }
}

// ---------------------------------------------------------------------------
// Pass 2: exact k-th-largest via 32-step MSB radix select on sortable keys,
// then write the binary node mask. One block, scores resident in LDS.
// ---------------------------------------------------------------------------
__global__ __launch_bounds__(1024) void topk_mask_kernel(const float* __restrict__ scores,
                                                         float* __restrict__ mask,
                                                         const int* __restrict__ topk_ptr) {
    constexpr int N = 8192;
    __shared__ unsigned keys[N];
    __shared__ int cnt_s;
    const int tid = threadIdx.x;
    const int k0  = topk_ptr[0];

    for (int i = tid; i < N; i += 1024) {
        unsigned u = __float_as_uint(scores[i]);
        keys[i] = (u & 0x80000000u) ? ~u : (u | 0x80000000u);   // order-preserving
    }
    __syncthreads();

    unsigned prefix = 0;
    int k = k0;
    for (int b = 31; b >= 0; --b) {
        if (tid == 0) cnt_s = 0;
        __syncthreads();
        const unsigned desired = prefix | (1u << b);
        const unsigned hmask   = ~((1u << b) - 1u);
        int local = 0;
        for (int i = tid; i < N; i += 1024)
            local += ((keys[i] & hmask) == desired);
        if (local) atomicAdd(&cnt_s, local);
        __syncthreads();
        const int c1 = cnt_s;
        if (k <= c1) prefix = desired; else k -= c1;   // uniform across threads
        __syncthreads();
    }
    // prefix == key of the k0-th largest score; scores are a.s. distinct
    for (int i = tid; i < N; i += 1024)
        mask[i] = (keys[i] >= prefix) ? 1.0f : 0.0f;
}

// ---------------------------------------------------------------------------
// Pass 4: subgraph[i][j] = graphs[i][j] * mask[j].  512 MB of streaming:
// b128 loads, non-temporal b128 stores, reversed walk to reuse the L2 tail
// left behind by pass 1. The 32 KB mask stays cache-hot.
// ---------------------------------------------------------------------------
__global__ __launch_bounds__(256) void subgraph_kernel(const float* __restrict__ graphs,
                                                       const float* __restrict__ mask,
                                                       float* __restrict__ out) {
    constexpr int N = 8192;
    const size_t total4 = (size_t)N * N / 4;          // 16M float4
    const size_t stride = (size_t)gridDim.x * blockDim.x;
    const v4f* g4 = reinterpret_cast<const v4f*>(graphs);
    const v4f* m4 = reinterpret_cast<const v4f*>(mask);
    v4f*       o4 = reinterpret_cast<v4f*>(out);

    for (size_t g = (size_t)blockIdx.x * blockDim.x + threadIdx.x;
         g < total4; g += stride) {
        const size_t idx = total4 - 1 - g;            // reverse: hit pass-1 L2 tail
        v4f a = g4[idx];
        v4f m = m4[idx & (size_t)(N / 4 - 1)];
        v4f r = a * m;
        __builtin_nontemporal_store(r, &o4[idx]);     // never re-read: NT store
    }
}

// ---------------------------------------------------------------------------
extern "C" void kernel_launch(void* const* d_in, const int* in_sizes, int n_in,
                              void* d_out, int out_size, void* d_ws, size_t ws_size,
                              hipStream_t stream) {
    constexpr int B = 1024, N = 8192;
    const float* x      = (const float*)d_in[0];   // [B, N]
    const float* graphs = (const float*)d_in[1];   // [N, N]
    const float* kern   = (const float*)d_in[2];   // [N, 1]
    const int*   topk   = (const int*)d_in[3];     // scalar (device)

    float* scores = (float*)d_ws;                  // [N]
    float* mask   = scores + N;                    // [N]

    float* xl_out  = (float*)d_out;                // [B] (== [B,1] flat)
    float* sub_out = xl_out + B;                   // [N, N]

    // s = graphs @ kernel  (WMMA GEMV)
    gemv16_wmma<<<N / 16, 256, 0, stream>>>(graphs, kern, scores, N);
    // threshold + binary node mask
    topk_mask_kernel<<<1, 1024, 0, stream>>>(scores, mask, topk);
    // xl = x @ s  (same WMMA GEMV; associativity of the reference matmuls)
    gemv16_wmma<<<B / 16, 256, 0, stream>>>(x, scores, xl_out, B);
    // subgraph = graphs * mask (broadcast over last axis)
    subgraph_kernel<<<32768, 256, 0, stream>>>(graphs, mask, sub_out);
}